// GraphBlock_10462540333121
// MI455X (gfx1250) — compile-verified
//
#include <hip/hip_runtime.h>
#include <hip/hip_bf16.h>
#include <math.h>

// ---------------------------------------------------------------------------
// CDNA5 (gfx1250) fp32 WMMA helpers: V_WMMA_F32_16X16X4_F32
//   A: 16x4 f32 -> 2 VGPR (v2f), B: 4x16 f32 -> 2 VGPR, C/D: 16x16 f32 -> v8f
// Lane layout (wave32):
//   A: m = lane&15, kk = (lane>>4)*2, holds A[m][k0+kk], A[m][k0+kk+1]
//   B: n = lane&15, kk = (lane>>4)*2, holds B[k0+kk][n], B[k0+kk+1][n]
//   C/D: n = lane&15, rows m = (lane>>4)*8 + r, r = 0..7
// Weights are pre-packed k-pair-interleaved so each lane's B fragment is one
// contiguous 8-byte load: Bp[((k>>1)*ldb + n)*2 + (k&1)] = B[k][n].
// ---------------------------------------------------------------------------
typedef float v2f __attribute__((ext_vector_type(2)));
typedef float v8f __attribute__((ext_vector_type(8)));

#define LDA_PAD 132   // 128+4: bank = 4*m + k  -> conflict-free fragment reads
#define LDA_PAD512 516

__device__ __forceinline__ void load_afrag_k128(v2f* areg, const float* sA,
                                                int mn, int kk) {
#pragma unroll
  for (int ks = 0; ks < 32; ++ks) {
    areg[ks][0] = sA[mn * LDA_PAD + ks * 4 + kk];
    areg[ks][1] = sA[mn * LDA_PAD + ks * 4 + kk + 1];
  }
}

// D(16x16) = A(16x128, register frags) * Bp(packed 128 x ldb)
__device__ __forceinline__ v8f wmma_k128p(const v2f* areg,
                                          const float* __restrict__ Bp,
                                          int ldb, int col0, int mn, int kk) {
  v8f acc = {};
#pragma unroll
  for (int ks = 0; ks < 32; ++ks) {
    const int kpair = (ks * 4 + kk) >> 1;  // k' = ks*4+kk is always even
    const v2f b =
        *(const v2f*)(Bp + ((size_t)kpair * ldb + col0 + mn) * 2);
    acc = __builtin_amdgcn_wmma_f32_16x16x4_f32(false, areg[ks], false, b,
                                                (short)0, acc, false, false);
  }
  return acc;
}

// LayerNorm of a 16 x 128 tile held in LDS (stride LDA_PAD), 256 threads,
// 16 threads per row, 8 elements each, butterfly reduce within 16-lane groups.
__device__ __forceinline__ void layernorm16(float* buf, int tid) {
  const int r = tid >> 4;
  const int g = tid & 15;
  float s = 0.f, s2 = 0.f;
#pragma unroll
  for (int j = 0; j < 8; ++j) {
    float v = buf[r * LDA_PAD + g * 8 + j];
    s += v; s2 += v * v;
  }
#pragma unroll
  for (int off = 1; off < 16; off <<= 1) {
    s  += __shfl_xor(s,  off, 32);
    s2 += __shfl_xor(s2, off, 32);
  }
  const float mean = s * (1.0f / 128.0f);
  const float var  = s2 * (1.0f / 128.0f) - mean * mean;
  const float rstd = rsqrtf(var + 1e-6f);
#pragma unroll
  for (int j = 0; j < 8; ++j) {
    const int idx = r * LDA_PAD + g * 8 + j;
    buf[idx] = (buf[idx] - mean) * rstd;
  }
}

__device__ __forceinline__ float silu_f(float x) {
  return x / (1.0f + __expf(-x));
}
__device__ __forceinline__ float gelu_tanh(float x) {
  const float t = 0.7978845608028654f * (x + 0.044715f * x * x * x);
  return 0.5f * x * (1.0f + tanhf(t));
}

// ---------------------------------------------------------------------------
// K0a: repack weight matrix [K x Ncols] into k-pair interleaved layout
// ---------------------------------------------------------------------------
__global__ void k_pack(const float* __restrict__ w, float* __restrict__ wp,
                       int K, int Ncols) {
  const int idx = blockIdx.x * 256 + threadIdx.x;
  if (idx >= K * Ncols) return;
  const int k = idx / Ncols, n2 = idx - k * Ncols;
  wp[((size_t)(k >> 1) * Ncols + n2) * 2 + (k & 1)] = w[idx];
}

// ---------------------------------------------------------------------------
// K0b: init scratch (smax = -inf, den = 0, agg = 0)
// ---------------------------------------------------------------------------
__global__ void k_init(float* __restrict__ smax, float* __restrict__ den,
                       float* __restrict__ agg, int n) {
  const int idx = blockIdx.x * 256 + threadIdx.x;
  if (idx < n * 8) { smax[idx] = -3.0e38f; den[idx] = 0.f; }
  if (idx < n * 128) agg[idx] = 0.f;
}

// ---------------------------------------------------------------------------
// K1: mods = silu(c) @ w_ada + b_ada  (16 x 768 tile per block)
//     sh_msa/sc_msa kept in LDS; g_msa/sh_mlp/sc_mlp/g_mlp -> mods512[N,512]
//     xm = LN(x)*(1+sc_msa)+sh_msa -> global
// ---------------------------------------------------------------------------
__global__ void k_ada(const float* __restrict__ x, const float* __restrict__ c,
                      const float* __restrict__ wp_ada,
                      const float* __restrict__ b_ada,
                      float* __restrict__ mods512, float* __restrict__ xm,
                      int n) {
  __shared__ float sA[16 * LDA_PAD];    // silu(c)
  __shared__ float sLn[16 * LDA_PAD];   // ln(x)
  __shared__ float sMod[16 * 256];      // sh_msa | sc_msa
  const int row0 = blockIdx.x * 16;
  const int tid = threadIdx.x;

  for (int i = tid; i < 16 * 128; i += 256) {
    const int r = i >> 7, j = i & 127;
    const float cv = c[(size_t)(row0 + r) * 128 + j];
    sA[r * LDA_PAD + j]  = silu_f(cv);
    sLn[r * LDA_PAD + j] = x[(size_t)(row0 + r) * 128 + j];
  }
  __syncthreads();
  layernorm16(sLn, tid);

  const int lane = tid & 31, wv = tid >> 5;
  const int mn = lane & 15, kk = (lane >> 4) << 1, mbase = (lane >> 4) * 8;
  v2f areg[32];
  load_afrag_k128(areg, sA, mn, kk);

#pragma unroll
  for (int i = 0; i < 6; ++i) {
    const int ct = wv * 6 + i;            // 48 col tiles / 8 waves
    const int col0 = ct * 16;
    v8f acc = wmma_k128p(areg, wp_ada, 768, col0, mn, kk);
    const float bias = b_ada[col0 + mn];
    if (col0 < 256) {
#pragma unroll
      for (int r = 0; r < 8; ++r)
        sMod[(mbase + r) * 256 + col0 + mn] = acc[r] + bias;
    } else {
#pragma unroll
      for (int r = 0; r < 8; ++r)
        mods512[(size_t)(row0 + mbase + r) * 512 + (col0 - 256) + mn] =
            acc[r] + bias;
    }
  }
  __syncthreads();

  for (int i = tid; i < 16 * 128; i += 256) {
    const int r = i >> 7, j = i & 127;
    const float ln = sLn[r * LDA_PAD + j];
    const float sh = sMod[r * 256 + j];
    const float sc = sMod[r * 256 + 128 + j];
    xm[(size_t)(row0 + r) * 128 + j] = ln * (1.0f + sc) + sh;
  }
}

// ---------------------------------------------------------------------------
// K2: qkv = xm @ w_qkv   [N,384]  (q cols 0..127, k 128..255, v 256..383)
// ---------------------------------------------------------------------------
__global__ void k_qkv(const float* __restrict__ xm,
                      const float* __restrict__ wp_qkv,
                      float* __restrict__ qkv, int n) {
  __shared__ float sA[16 * LDA_PAD];
  const int row0 = blockIdx.x * 16;
  const int tid = threadIdx.x;
  for (int i = tid; i < 16 * 128; i += 256) {
    const int r = i >> 7, j = i & 127;
    sA[r * LDA_PAD + j] = xm[(size_t)(row0 + r) * 128 + j];
  }
  __syncthreads();
  const int lane = tid & 31, wv = tid >> 5;
  const int mn = lane & 15, kk = (lane >> 4) << 1, mbase = (lane >> 4) * 8;
  v2f areg[32];
  load_afrag_k128(areg, sA, mn, kk);
#pragma unroll
  for (int i = 0; i < 3; ++i) {
    const int col0 = (wv * 3 + i) * 16;   // 24 col tiles / 8 waves
    v8f acc = wmma_k128p(areg, wp_qkv, 384, col0, mn, kk);
#pragma unroll
    for (int r = 0; r < 8; ++r)
      qkv[(size_t)(row0 + mbase + r) * 384 + col0 + mn] = acc[r];
  }
}

// ---------------------------------------------------------------------------
// K3: per-edge score + segment max (float atomicMax into L2)
// ---------------------------------------------------------------------------
__global__ void k_score(const float* __restrict__ qkv,
                        const int* __restrict__ ei,
                        float* __restrict__ score, float* __restrict__ smax,
                        int e) {
  const int idx = blockIdx.x * 256 + threadIdx.x;
  if (idx >= e * 8) return;
  const int eid = idx >> 3, h = idx & 7;
  const int j = ei[eid];        // src
  const int i = ei[e + eid];    // dst
  const float4* qp = (const float4*)(qkv + (size_t)i * 384 + h * 16);
  const float4* kp = (const float4*)(qkv + (size_t)j * 384 + 128 + h * 16);
  float s = 0.f;
#pragma unroll
  for (int t = 0; t < 4; ++t) {
    const float4 a = qp[t], b = kp[t];
    s += a.x * b.x + a.y * b.y + a.z * b.z + a.w * b.w;
  }
  s *= 0.25f;                    // 1/sqrt(16)
  score[idx] = s;
  atomicMax(&smax[i * 8 + h], s);
}

// ---------------------------------------------------------------------------
// K4: ex = exp(score - smax[dst]); den += ex; agg += v[src]*ex (unnormalized)
//     one thread per (edge, float4 of the 128 message dims)
// ---------------------------------------------------------------------------
__global__ void k_aggregate(const float* __restrict__ qkv,
                            const int* __restrict__ ei,
                            const float* __restrict__ score,
                            const float* __restrict__ smax,
                            float* __restrict__ den, float* __restrict__ agg,
                            int e) {
  const long long idx = (long long)blockIdx.x * 256 + threadIdx.x;
  if (idx >= (long long)e * 32) return;
  const int eid = (int)(idx >> 5);
  const int d = (int)(idx & 31) * 4;     // 0,4,...,124
  const int h = d >> 4;
  const int j = ei[eid];
  const int i = ei[e + eid];
  const float s = score[eid * 8 + h];
  const float m = smax[i * 8 + h];
  const float ex = __expf(s - m);
  if ((d & 15) == 0) atomicAdd(&den[i * 8 + h], ex);
  const float4 v = *(const float4*)(qkv + (size_t)j * 384 + 256 + d);
  float* ap = agg + (size_t)i * 128 + d;
  atomicAdd(ap + 0, v.x * ex);
  atomicAdd(ap + 1, v.y * ex);
  atomicAdd(ap + 2, v.z * ex);
  atomicAdd(ap + 3, v.w * ex);
}

// ---------------------------------------------------------------------------
// K5: aout = (agg/den) @ w_proj + b_proj ; x1 = x + g_msa*aout
//     hin = LN(x1)*(1+sc_mlp)+sh_mlp
// ---------------------------------------------------------------------------
__global__ void k_proj(const float* __restrict__ x,
                       const float* __restrict__ agg,
                       const float* __restrict__ den,
                       const float* __restrict__ wp_proj,
                       const float* __restrict__ b_proj,
                       const float* __restrict__ mods512,
                       float* __restrict__ x1, float* __restrict__ hin,
                       int n) {
  __shared__ float sA[16 * LDA_PAD];
  __shared__ float sX[16 * LDA_PAD];
  const int row0 = blockIdx.x * 16;
  const int tid = threadIdx.x;
  for (int i = tid; i < 16 * 128; i += 256) {
    const int r = i >> 7, d = i & 127;
    const float dn = den[(row0 + r) * 8 + (d >> 4)];
    const float a = agg[(size_t)(row0 + r) * 128 + d];
    sA[r * LDA_PAD + d] = (dn > 0.f) ? a / dn : 0.f;
  }
  __syncthreads();
  const int lane = tid & 31, wv = tid >> 5;
  const int mn = lane & 15, kk = (lane >> 4) << 1, mbase = (lane >> 4) * 8;
  v2f areg[32];
  load_afrag_k128(areg, sA, mn, kk);
  const int col0 = wv * 16;                 // 8 col tiles / 8 waves
  v8f acc = wmma_k128p(areg, wp_proj, 128, col0, mn, kk);
  const float bias = b_proj[col0 + mn];
#pragma unroll
  for (int r = 0; r < 8; ++r) {
    const int m = mbase + r;
    const size_t gr = (size_t)(row0 + m);
    const float g = mods512[gr * 512 + 0 + col0 + mn];      // g_msa
    const float v = x[gr * 128 + col0 + mn] + g * (acc[r] + bias);
    sX[m * LDA_PAD + col0 + mn] = v;
    x1[gr * 128 + col0 + mn] = v;
  }
  __syncthreads();
  layernorm16(sX, tid);
  __syncthreads();
  for (int i = tid; i < 16 * 128; i += 256) {
    const int r = i >> 7, j = i & 127;
    const size_t gr = (size_t)(row0 + r);
    const float ln = sX[r * LDA_PAD + j];
    const float sh = mods512[gr * 512 + 128 + j];           // sh_mlp
    const float sc = mods512[gr * 512 + 256 + j];           // sc_mlp
    hin[gr * 128 + j] = ln * (1.0f + sc) + sh;
  }
}

// ---------------------------------------------------------------------------
// K6: hmid = gelu(hin @ w_mlp1 + b_mlp1)   [N,512]
// ---------------------------------------------------------------------------
__global__ void k_mlp1(const float* __restrict__ hin,
                       const float* __restrict__ wp1,
                       const float* __restrict__ b1,
                       float* __restrict__ hmid, int n) {
  __shared__ float sA[16 * LDA_PAD];
  const int row0 = blockIdx.x * 16;
  const int tid = threadIdx.x;
  for (int i = tid; i < 16 * 128; i += 256) {
    const int r = i >> 7, j = i & 127;
    sA[r * LDA_PAD + j] = hin[(size_t)(row0 + r) * 128 + j];
  }
  __syncthreads();
  const int lane = tid & 31, wv = tid >> 5;
  const int mn = lane & 15, kk = (lane >> 4) << 1, mbase = (lane >> 4) * 8;
  v2f areg[32];
  load_afrag_k128(areg, sA, mn, kk);
#pragma unroll
  for (int i = 0; i < 4; ++i) {
    const int col0 = (wv * 4 + i) * 16;     // 32 col tiles / 8 waves
    v8f acc = wmma_k128p(areg, wp1, 512, col0, mn, kk);
    const float bias = b1[col0 + mn];
#pragma unroll
    for (int r = 0; r < 8; ++r) {
      const float u = acc[r] + bias;
      hmid[(size_t)(row0 + mbase + r) * 512 + col0 + mn] = gelu_tanh(u);
    }
  }
}

// ---------------------------------------------------------------------------
// K7: out = x1 + g_mlp * (hmid @ w_mlp2 + b_mlp2)
//     hmid tile (32 KB) staged into LDS with CDNA5 async global->LDS DMA
// ---------------------------------------------------------------------------
__global__ void k_mlp2(const float* __restrict__ hmid,
                       const float* __restrict__ wp2,
                       const float* __restrict__ b2,
                       const float* __restrict__ x1,
                       const float* __restrict__ mods512,
                       float* __restrict__ out, int n) {
  __shared__ float sA[16 * LDA_PAD512];     // 16 x 512 tile, padded
  const int row0 = blockIdx.x * 16;
  const int tid = threadIdx.x;

  // Async copy 16x512 floats (2048 x b128) global -> LDS, ASYNCcnt-tracked.
#pragma unroll
  for (int t = 0; t < 8; ++t) {
    const int cidx = tid + t * 256;         // b128 chunk index
    const int r = cidx >> 7, jc = cidx & 127;
    const unsigned ldsa =
        (unsigned)(size_t)(&sA[r * LDA_PAD512 + jc * 4]);
    const unsigned long long ga =
        (unsigned long long)(hmid + (size_t)(row0 + r) * 512 + jc * 4);
    asm volatile("global_load_async_to_lds_b128 %0, %1, off"
                 :: "v"(ldsa), "v"(ga) : "memory");
  }
  asm volatile("s_wait_asynccnt 0x0" ::: "memory");
  __syncthreads();

  const int lane = tid & 31, wv = tid >> 5;
  const int mn = lane & 15, kk = (lane >> 4) << 1, mbase = (lane >> 4) * 8;
  const int col0 = wv * 16;                 // 8 col tiles / 8 waves
  v8f acc = {};
#pragma unroll
  for (int ks = 0; ks < 128; ++ks) {        // K = 512
    v2f a;
    a[0] = sA[mn * LDA_PAD512 + ks * 4 + kk];
    a[1] = sA[mn * LDA_PAD512 + ks * 4 + kk + 1];
    const int kpair = (ks * 4 + kk) >> 1;
    const v2f b = *(const v2f*)(wp2 + ((size_t)kpair * 128 + col0 + mn) * 2);
    acc = __builtin_amdgcn_wmma_f32_16x16x4_f32(false, a, false, b, (short)0,
                                                acc, false, false);
  }
  const float bias = b2[col0 + mn];
#pragma unroll
  for (int r = 0; r < 8; ++r) {
    const size_t gr = (size_t)(row0 + mbase + r);
    const float g = mods512[gr * 512 + 384 + col0 + mn];    // g_mlp
    out[gr * 128 + col0 + mn] =
        x1[gr * 128 + col0 + mn] + g * (acc[r] + bias);
  }
}

// ---------------------------------------------------------------------------
extern "C" void kernel_launch(void* const* d_in, const int* in_sizes, int n_in,
                              void* d_out, int out_size, void* d_ws,
                              size_t ws_size, hipStream_t stream) {
  const float* x      = (const float*)d_in[0];
  const int*   ei     = (const int*)d_in[1];
  const float* c      = (const float*)d_in[2];
  const float* w_qkv  = (const float*)d_in[3];
  const float* w_proj = (const float*)d_in[4];
  const float* b_proj = (const float*)d_in[5];
  const float* w_mlp1 = (const float*)d_in[6];
  const float* b_mlp1 = (const float*)d_in[7];
  const float* w_mlp2 = (const float*)d_in[8];
  const float* b_mlp2 = (const float*)d_in[9];
  const float* w_ada  = (const float*)d_in[10];
  const float* b_ada  = (const float*)d_in[11];

  const int n = in_sizes[0] / 128;   // 50000
  const int e = in_sizes[1] / 2;     // 800000

  float* ws    = (float*)d_ws;
  float* mods  = ws;                            // [n,512] g_msa|sh_mlp|sc_mlp|g_mlp
  float* xm    = mods + (size_t)n * 512;        // [n,128] (modulated x, later hin)
  float* qkv   = xm + (size_t)n * 128;          // [n,384]
  float* score = qkv + (size_t)n * 384;         // [e,8]
  float* smax  = score + (size_t)e * 8;         // [n,8]
  float* den   = smax + (size_t)n * 8;          // [n,8]
  float* agg   = den + (size_t)n * 8;           // [n,128]
  float* x1    = agg + (size_t)n * 128;         // [n,128]
  float* hmid  = x1 + (size_t)n * 128;          // [n,512]
  float* wp_ada  = hmid + (size_t)n * 512;      // packed weights (1.2 MB total)
  float* wp_qkv  = wp_ada + 128 * 768;
  float* wp_proj = wp_qkv + 128 * 384;
  float* wp1     = wp_proj + 128 * 128;
  float* wp2     = wp1 + 128 * 512;
  float* hin   = xm;                            // reuse xm after K2 consumes it
  float* out   = (float*)d_out;

  const int ntiles = n / 16;                    // 3125

  k_pack<<<(128 * 768 + 255) / 256, 256, 0, stream>>>(w_ada, wp_ada, 128, 768);
  k_pack<<<(128 * 384 + 255) / 256, 256, 0, stream>>>(w_qkv, wp_qkv, 128, 384);
  k_pack<<<(128 * 128 + 255) / 256, 256, 0, stream>>>(w_proj, wp_proj, 128, 128);
  k_pack<<<(128 * 512 + 255) / 256, 256, 0, stream>>>(w_mlp1, wp1, 128, 512);
  k_pack<<<(512 * 128 + 255) / 256, 256, 0, stream>>>(w_mlp2, wp2, 512, 128);
  k_init<<<(n * 128 + 255) / 256, 256, 0, stream>>>(smax, den, agg, n);

  k_ada<<<ntiles, 256, 0, stream>>>(x, c, wp_ada, b_ada, mods, xm, n);
  k_qkv<<<ntiles, 256, 0, stream>>>(xm, wp_qkv, qkv, n);
  k_score<<<(e * 8 + 255) / 256, 256, 0, stream>>>(qkv, ei, score, smax, e);
  {
    const long long tot = (long long)e * 32;
    k_aggregate<<<(int)((tot + 255) / 256), 256, 0, stream>>>(qkv, ei, score,
                                                              smax, den, agg, e);
  }
  k_proj<<<ntiles, 256, 0, stream>>>(x, agg, den, wp_proj, b_proj, mods, x1,
                                     hin, n);
  k_mlp1<<<ntiles, 256, 0, stream>>>(hin, wp1, b_mlp1, hmid, n);
  k_mlp2<<<ntiles, 256, 0, stream>>>(hmid, wp2, b_mlp2, x1, mods, out, n);
}